// RankGNN_16381005267206
// MI455X (gfx1250) — compile-verified
//
#include <hip/hip_runtime.h>
#include <hip/hip_bf16.h>
#include <math.h>

#define N_NODES 50000
#define N_EDGES 600000
#define N_GRAPH 256
#define F_IN    3
#define F_H     128
#define F_OUT   10
#define LDS_ROW 132   // padded LDS row stride in floats (528 B: 16B-aligned, bank-spread)

typedef __attribute__((ext_vector_type(2))) float v2f;
typedef __attribute__((ext_vector_type(8))) float v8f;

// ---------------- small utility kernels ----------------

__global__ void k_set1(float* p, int n) {
    int i = blockIdx.x * blockDim.x + threadIdx.x;
    if (i < n) p[i] = 1.0f;          // self-loop contributes weight 1 to degree
}

__global__ void k_zero(float* p, int n) {
    int i = blockIdx.x * blockDim.x + threadIdx.x;
    if (i < n) p[i] = 0.0f;
}

__global__ void k_deg(const int* __restrict__ dst, const float* __restrict__ ew,
                      float* __restrict__ deg, int e) {
    int i = blockIdx.x * blockDim.x + threadIdx.x;
    if (i < e) atomicAdd(&deg[dst[i]], ew[i]);
}

__global__ void k_dinv(const float* __restrict__ deg, float* __restrict__ dinv,
                       float* __restrict__ selfn, int n) {
    int i = blockIdx.x * blockDim.x + threadIdx.x;
    if (i < n) {
        float d = deg[i];
        float r = (d > 0.0f) ? rsqrtf(fmaxf(d, 1e-12f)) : 0.0f;
        dinv[i]  = r;
        selfn[i] = r * r;            // self-loop norm: dinv[i]*1*dinv[i]
    }
}

__global__ void k_norm(const int* __restrict__ src, const int* __restrict__ dst,
                       const float* __restrict__ ew, const float* __restrict__ dinv,
                       float* __restrict__ norm, int e) {
    int i = blockIdx.x * blockDim.x + threadIdx.x;
    if (i < e) norm[i] = dinv[src[i]] * ew[i] * dinv[dst[i]];
}

// ---------------- layer-1 GEMM (K=3, trivial) ----------------

__global__ void k_gemm_in3(const float* __restrict__ x, const float* __restrict__ W,
                           float* __restrict__ h, int n) {
    int i = blockIdx.x * blockDim.x + threadIdx.x;
    if (i < n * F_H) {
        int r = i / F_H, c = i % F_H;
        const float* xr = x + (size_t)r * F_IN;
        h[i] = xr[0] * W[0 * F_H + c] + xr[1] * W[1 * F_H + c] + xr[2] * W[2 * F_H + c];
    }
}

// ---------------- WMMA fp32 GEMM: C[M,128] = op(A)[M,128] * B[128,128] ----------------
// One block (8 waves) per 16-row tile; wave w owns column tile w. The shared
// 16x128 A tile is staged into LDS once per block with async global->LDS DMA
// (ASYNCcnt path), then each wave runs 32 x V_WMMA_F32_16X16X4_F32 reading A
// from LDS and B (weights, L2/WGP$-resident) from global. Epilogue/prologue
// fusion (ReLU on A, bias+ReLU on store) resolved at compile time.

template <bool RELU_A, bool HAS_BIAS, bool RELU_S>
__global__ void __launch_bounds__(256)
k_gemm_wmma(const float* __restrict__ A, const float* __restrict__ B,
            float* __restrict__ C, const float* __restrict__ bias, int M) {
    __shared__ __align__(16) float As[16 * LDS_ROW];

    int rowTile = blockIdx.x;                  // block-uniform: barrier-safe guard
    if (rowTile * 16 >= M) return;
    int tid = threadIdx.x;

    // ---- async stage: 16 rows x 128 f32 (contiguous 8 KB) -> padded LDS rows ----
    {
        unsigned long long gbase =
            (unsigned long long)(A + (size_t)rowTile * 16 * F_H);
#pragma unroll
        for (int it = 0; it < 2; ++it) {
            int chunk = tid + it * 256;        // 512 chunks of 16 B
            int r = chunk >> 5;                // row 0..15
            int c = chunk & 31;                // 16B-chunk within row
            unsigned lds  = (unsigned)(unsigned long long)(&As[r * LDS_ROW]) + c * 16;
            unsigned goff = (unsigned)(r * (F_H * 4) + c * 16);
            asm volatile("global_load_async_to_lds_b128 %0, %1, %2"
                         :: "v"(lds), "v"(goff), "s"(gbase) : "memory");
        }
        asm volatile("s_wait_asynccnt 0" ::: "memory");
    }
    __syncthreads();

    int colTile = tid >> 5;                    // wave id = column tile (0..7)
    int lane = tid & 31;
    int half = lane >> 4;                      // 0: lanes 0-15, 1: lanes 16-31
    int l16  = lane & 15;
    int col  = colTile * 16 + l16;             // B/C/D: N = lane%16

    v8f acc = {};
#pragma unroll
    for (int k = 0; k < F_H; k += 4) {
        int ka = k + half * 2;                 // K split by lane-half (ISA 16x4 layout)
        float2 a2 = *(const float2*)&As[l16 * LDS_ROW + ka];   // ds_load_b64
        float a0 = a2.x, a1 = a2.y;
        if (RELU_A) { a0 = fmaxf(a0, 0.0f); a1 = fmaxf(a1, 0.0f); }
        v2f av; av.x = a0; av.y = a1;
        v2f bv;
        bv.x = B[(size_t)ka * F_H + col];
        bv.y = B[(size_t)(ka + 1) * F_H + col];
        acc = __builtin_amdgcn_wmma_f32_16x16x4_f32(
            /*neg_a=*/false, av, /*neg_b=*/false, bv,
            /*c_mod=*/(short)0, acc, /*reuse_a=*/false, /*reuse_b=*/false);
    }

    float bcol = HAS_BIAS ? bias[col] : 0.0f;
#pragma unroll
    for (int i = 0; i < 8; ++i) {              // C/D: M = vgpr + 8*half
        int r = rowTile * 16 + half * 8 + i;
        float v = acc[i];
        if (HAS_BIAS) v += bcol;
        if (RELU_S)   v = fmaxf(v, 0.0f);
        if (r < M) C[(size_t)r * F_H + col] = v;
    }
}

// ---------------- aggregation ----------------

// y[i][c] = b[c] + selfnorm[i] * h[i][c]   (bias + self-loop message)
__global__ void k_init_y(const float* __restrict__ h, const float* __restrict__ selfn,
                         const float* __restrict__ b, float* __restrict__ y, int n) {
    int i = blockIdx.x * blockDim.x + threadIdx.x;
    if (i < n * F_H) {
        int r = i / F_H, c = i % F_H;
        y[i] = b[c] + selfn[r] * h[i];
    }
}

// one wave per edge: y[dst] += norm[e] * h[src]; 4 floats per lane (32*4 = 128)
__global__ void k_scatter(const int* __restrict__ src, const int* __restrict__ dst,
                          const float* __restrict__ norm, const float* __restrict__ h,
                          float* __restrict__ y, int e) {
    int wave = (blockIdx.x * blockDim.x + threadIdx.x) >> 5;
    int lane = threadIdx.x & 31;
    if (wave >= e) return;
    int   s = src[wave];
    int   d = dst[wave];
    float w = norm[wave];
    float4 m = reinterpret_cast<const float4*>(h + (size_t)s * F_H)[lane];
    float* yr = y + (size_t)d * F_H + lane * 4;
    atomicAdd(yr + 0, w * m.x);
    atomicAdd(yr + 1, w * m.y);
    atomicAdd(yr + 2, w * m.z);
    atomicAdd(yr + 3, w * m.w);
}

// one wave per node: segment-sum of relu(y) into per-graph buckets
__global__ void k_pool(const float* __restrict__ y, const int* __restrict__ batch,
                       float* __restrict__ sums, float* __restrict__ cnt, int n) {
    int wave = (blockIdx.x * blockDim.x + threadIdx.x) >> 5;
    int lane = threadIdx.x & 31;
    if (wave >= n) return;
    int g = batch[wave];
    float4 m = reinterpret_cast<const float4*>(y + (size_t)wave * F_H)[lane];
    float* sr = sums + (size_t)g * F_H + lane * 4;
    atomicAdd(sr + 0, fmaxf(m.x, 0.0f));
    atomicAdd(sr + 1, fmaxf(m.y, 0.0f));
    atomicAdd(sr + 2, fmaxf(m.z, 0.0f));
    atomicAdd(sr + 3, fmaxf(m.w, 0.0f));
    if (lane == 0) atomicAdd(&cnt[g], 1.0f);
}

__global__ void k_mean(const float* __restrict__ sums, const float* __restrict__ cnt,
                       float* __restrict__ gemb, int g) {
    int i = blockIdx.x * blockDim.x + threadIdx.x;
    if (i < g * F_H) gemb[i] = sums[i] / fmaxf(cnt[i / F_H], 1.0f);
}

// logits = hc @ Wc2 + bc2 ; stability = sigmoid(gemb @ Wa + ba)
__global__ void k_head(const float* __restrict__ gemb, const float* __restrict__ hc,
                       const float* __restrict__ Wc2, const float* __restrict__ bc2,
                       const float* __restrict__ Wa, const float* __restrict__ ba,
                       float* __restrict__ out) {
    int t = blockIdx.x * blockDim.x + threadIdx.x;
    if (t < N_GRAPH * F_OUT) {
        int g = t / F_OUT, o = t - g * F_OUT;
        float acc = bc2[o];
        const float* hr = hc + (size_t)g * F_H;
        for (int k = 0; k < F_H; ++k) acc += hr[k] * Wc2[(size_t)k * F_OUT + o];
        out[t] = acc;
    } else if (t < N_GRAPH * F_OUT + N_GRAPH) {
        int g = t - N_GRAPH * F_OUT;
        float z = ba[0];
        const float* gr = gemb + (size_t)g * F_H;
        for (int k = 0; k < F_H; ++k) z += gr[k] * Wa[k];
        out[t] = 1.0f / (1.0f + expf(-z));
    }
}

// ---------------- launcher ----------------

extern "C" void kernel_launch(void* const* d_in, const int* in_sizes, int n_in,
                              void* d_out, int out_size, void* d_ws, size_t ws_size,
                              hipStream_t stream) {
    (void)in_sizes; (void)n_in; (void)out_size; (void)ws_size;

    const float* x   = (const float*)d_in[0];
    const int*   ei  = (const int*)d_in[1];
    const int*   bat = (const int*)d_in[2];
    const float* ew  = (const float*)d_in[3];
    const float* W1  = (const float*)d_in[4];  const float* b1  = (const float*)d_in[5];
    const float* W2  = (const float*)d_in[6];  const float* b2  = (const float*)d_in[7];
    const float* W3  = (const float*)d_in[8];  const float* b3  = (const float*)d_in[9];
    const float* W4  = (const float*)d_in[10]; const float* b4  = (const float*)d_in[11];
    const float* Wc1 = (const float*)d_in[12]; const float* bc1 = (const float*)d_in[13];
    const float* Wc2 = (const float*)d_in[14]; const float* bc2 = (const float*)d_in[15];
    const float* Wa  = (const float*)d_in[16]; const float* ba  = (const float*)d_in[17];
    const int* src = ei;
    const int* dst = ei + N_EDGES;
    float* out = (float*)d_out;

    // workspace carve-up (offsets padded to 256 floats = 1 KB)
    float* ws = (float*)d_ws;
    size_t off = 0;
    auto carve = [&](size_t n) { float* p = ws + off; off += (n + 255) & ~(size_t)255; return p; };
    float* deg   = carve(N_NODES);
    float* dinv  = carve(N_NODES);
    float* selfn = carve(N_NODES);
    float* norm  = carve(N_EDGES);
    float* h     = carve((size_t)N_NODES * F_H);
    float* yA    = carve((size_t)N_NODES * F_H);
    float* yB    = carve((size_t)N_NODES * F_H);
    float* sums  = carve((size_t)N_GRAPH * F_H);
    float* cnt   = carve(N_GRAPH);
    float* gemb  = carve((size_t)N_GRAPH * F_H);
    float* hc    = carve((size_t)N_GRAPH * F_H);

    const int T = 256;
    const int NH = N_NODES * F_H;

    // GCN normalization
    k_set1<<<(N_NODES + T - 1) / T, T, 0, stream>>>(deg, N_NODES);
    k_deg <<<(N_EDGES + T - 1) / T, T, 0, stream>>>(dst, ew, deg, N_EDGES);
    k_dinv<<<(N_NODES + T - 1) / T, T, 0, stream>>>(deg, dinv, selfn, N_NODES);
    k_norm<<<(N_EDGES + T - 1) / T, T, 0, stream>>>(src, dst, ew, dinv, norm, N_EDGES);

    // layer 1 (K=3 scalar GEMM)
    k_gemm_in3<<<(NH + T - 1) / T, T, 0, stream>>>(x, W1, h, N_NODES);
    k_init_y  <<<(NH + T - 1) / T, T, 0, stream>>>(h, selfn, b1, yA, N_NODES);
    k_scatter <<<(N_EDGES * 32 + T - 1) / T, T, 0, stream>>>(src, dst, norm, h, yA, N_EDGES);

    // layers 2-4: WMMA GEMM (async-LDS staged A, ReLU fused on A-load)
    const float* Ws[3] = {W2, W3, W4};
    const float* bs[3] = {b2, b3, b4};
    float* ycur = yA;
    float* ynxt = yB;
    const int gemmBlocks = N_NODES / 16;                 // 3125 blocks, 8 waves each
    for (int l = 0; l < 3; ++l) {
        k_gemm_wmma<true, false, false><<<gemmBlocks, T, 0, stream>>>(
            ycur, Ws[l], h, nullptr, N_NODES);
        k_init_y <<<(NH + T - 1) / T, T, 0, stream>>>(h, selfn, bs[l], ynxt, N_NODES);
        k_scatter<<<(N_EDGES * 32 + T - 1) / T, T, 0, stream>>>(src, dst, norm, h, ynxt, N_EDGES);
        float* t = ycur; ycur = ynxt; ynxt = t;
    }

    // global mean pool (ReLU of final layer fused on load)
    k_zero<<<(N_GRAPH * F_H + T - 1) / T, T, 0, stream>>>(sums, N_GRAPH * F_H);
    k_zero<<<(N_GRAPH + T - 1) / T, T, 0, stream>>>(cnt, N_GRAPH);
    k_pool<<<(N_NODES * 32 + T - 1) / T, T, 0, stream>>>(ycur, bat, sums, cnt, N_NODES);
    k_mean<<<(N_GRAPH * F_H + T - 1) / T, T, 0, stream>>>(sums, cnt, gemb, N_GRAPH);

    // head: hc = relu(gemb @ Wc1 + bc1) via WMMA (bias+ReLU fused on store)
    k_gemm_wmma<false, true, true><<<N_GRAPH / 16, T, 0, stream>>>(
        gemb, Wc1, hc, bc1, N_GRAPH);

    // logits + stability -> d_out
    k_head<<<(N_GRAPH * (F_OUT + 1) + T - 1) / T, T, 0, stream>>>(
        gemb, hc, Wc2, bc2, Wa, ba, out);
}